// NeuralHSMM_9749575762408
// MI455X (gfx1250) — compile-verified
//
#include <hip/hip_runtime.h>
#include <math.h>

// Problem constants (match reference)
#define KST   128      // n_states
#define DMAX  48       // max duration
#define NF    256      // n_features
#define CDIM  256      // context dim
#define BATCH 16
#define TLEN  768
#define NEGV  (-1e9f)
#define LOG2PI_F 1.8378770664093453f
#define MIN_COVAR_F 1e-3f

typedef float v2f __attribute__((ext_vector_type(2)));
typedef float v8f __attribute__((ext_vector_type(8)));

// ---------------------------------------------------------------------------
// Kernel 1: A = softmax(trans_logits + 0.1*tanh(ctx_A_w@ctx + b)) rows.
// (scan uses exp(logA) == softmax directly)
// ---------------------------------------------------------------------------
__global__ __launch_bounds__(KST) void ctx_transA_kernel(
    const float* __restrict__ trans_logits, const float* __restrict__ w,
    const float* __restrict__ bias, const float* __restrict__ ctx,
    float* __restrict__ Amat) {
  __shared__ float cs[CDIM];
  __shared__ float red[KST];
  const int i = blockIdx.x, j = threadIdx.x;
  for (int c = j; c < CDIM; c += KST) cs[c] = ctx[c];
  __syncthreads();
  const int row = i * KST + j;
  const float* wr = w + (size_t)row * CDIM;
  float acc = 0.f;
#pragma unroll 8
  for (int c = 0; c < CDIM; ++c) acc += wr[c] * cs[c];
  const float u = trans_logits[row] + 0.1f * tanhf(acc + bias[row]);
  red[j] = u; __syncthreads();
  for (int off = 64; off > 0; off >>= 1) { if (j < off) red[j] = fmaxf(red[j], red[j + off]); __syncthreads(); }
  const float mx = red[0]; __syncthreads();
  const float e = expf(u - mx);
  red[j] = e; __syncthreads();
  for (int off = 64; off > 0; off >>= 1) { if (j < off) red[j] += red[j + off]; __syncthreads(); }
  Amat[row] = e / red[0];
}

// ---------------------------------------------------------------------------
// Kernel 2: logDT[d][k] = log_softmax_d(dur_logits + 0.1*tanh(ctx_D_w@ctx+b))
// stored transposed (DMAX x KST).
// ---------------------------------------------------------------------------
__global__ __launch_bounds__(64) void ctx_durD_kernel(
    const float* __restrict__ dur_logits, const float* __restrict__ w,
    const float* __restrict__ bias, const float* __restrict__ ctx,
    float* __restrict__ logDT) {
  __shared__ float cs[CDIM];
  __shared__ float red[64];
  const int k = blockIdx.x, j = threadIdx.x;
  for (int c = j; c < CDIM; c += 64) cs[c] = ctx[c];
  __syncthreads();
  float u = NEGV;
  if (j < DMAX) {
    const int row = k * DMAX + j;
    const float* wr = w + (size_t)row * CDIM;
    float acc = 0.f;
#pragma unroll 8
    for (int c = 0; c < CDIM; ++c) acc += wr[c] * cs[c];
    u = dur_logits[row] + 0.1f * tanhf(acc + bias[row]);
  }
  red[j] = u; __syncthreads();
  for (int off = 32; off > 0; off >>= 1) { if (j < off) red[j] = fmaxf(red[j], red[j + off]); __syncthreads(); }
  const float mx = red[0]; __syncthreads();
  const float e = (j < DMAX) ? expf(u - mx) : 0.f;
  red[j] = e; __syncthreads();
  for (int off = 32; off > 0; off >>= 1) { if (j < off) red[j] += red[j + off]; __syncthreads(); }
  const float ls = logf(red[0]);
  if (j < DMAX) logDT[j * KST + k] = u - mx - ls;
}

// ---------------------------------------------------------------------------
// Kernel 3: emission GEMM operands.
//   Bmat[f][k]    = -0.5 * inv[k][f]
//   Bmat[NF+f][k] =  mu_eff[k][f] * inv[k][f]
//   dvec[k]       = -0.5*(sum_f mu^2*inv + sum_f log var + NF*LOG2PI)
// ---------------------------------------------------------------------------
__global__ __launch_bounds__(NF) void ctx_emis_kernel(
    const float* __restrict__ mu, const float* __restrict__ log_var,
    const float* __restrict__ w, const float* __restrict__ bias,
    const float* __restrict__ ctx, float* __restrict__ Bmat,
    float* __restrict__ dvec) {
  __shared__ float cs[CDIM];
  __shared__ float red[NF];
  const int k = blockIdx.x, f = threadIdx.x;
  cs[f] = ctx[f]; __syncthreads();
  const int row = k * NF + f;
  const float* wr = w + (size_t)row * CDIM;
  float acc = 0.f;
#pragma unroll 8
  for (int c = 0; c < CDIM; ++c) acc += wr[c] * cs[c];
  const float me = mu[row] + 0.1f * (acc + bias[row]);
  const float lv = log_var[row];
  const float sp = (lv > 20.f) ? lv : log1pf(expf(lv));
  const float var = sp + MIN_COVAR_F;
  const float iv = 1.f / var;
  Bmat[f * KST + k] = -0.5f * iv;
  Bmat[(NF + f) * KST + k] = me * iv;
  red[f] = me * me * iv + logf(var);
  __syncthreads();
  for (int off = 128; off > 0; off >>= 1) { if (f < off) red[f] += red[f + off]; __syncthreads(); }
  if (f == 0) dvec[k] = -0.5f * (red[0] + (float)NF * LOG2PI_F);
}

// ---------------------------------------------------------------------------
// Kernel 4: logpi = log_softmax(pi_logits).
// ---------------------------------------------------------------------------
__global__ __launch_bounds__(KST) void pi_kernel(const float* __restrict__ pl,
                                                 float* __restrict__ lpi) {
  __shared__ float red[KST];
  const int j = threadIdx.x;
  const float u = pl[j];
  red[j] = u; __syncthreads();
  for (int off = 64; off > 0; off >>= 1) { if (j < off) red[j] = fmaxf(red[j], red[j + off]); __syncthreads(); }
  const float mx = red[0]; __syncthreads();
  red[j] = expf(u - mx); __syncthreads();
  for (int off = 64; off > 0; off >>= 1) { if (j < off) red[j] += red[j + off]; __syncthreads(); }
  lpi[j] = u - mx - logf(red[0]);
}

// ---------------------------------------------------------------------------
// Kernel 5: emission log-probs via fp32 WMMA (v_wmma_f32_16x16x4_f32).
//   logb(12288 x 128) = [x^2 | x](12288 x 512) @ Bmat(512 x 128) + dvec
// Block = 32 rows (two 16-row M tiles, dual accumulators sharing each B
// fragment); 8 waves each own one 16-col N tile.
// Branch-free: kk-loop split statically at the x^2 / x boundary (kk==64).
// Fragment layout per ISA: A lanes0-15 = {K0,K1}, lanes16-31 = {K2,K3};
// B mirrored; C/D VGPR r holds rows M=r (lanes0-15) and M=r+8 (lanes16-31).
// ---------------------------------------------------------------------------
__global__ __launch_bounds__(256) void emis_gemm_kernel(
    const float* __restrict__ x, const float* __restrict__ Bmat,
    const float* __restrict__ dvec, float* __restrict__ logb) {
  __shared__ float xs[32][258];   // padded (even) for bank spread + b64 align
  const int mt = blockIdx.x;      // 0..383 : rows 32*mt .. 32*mt+31
  const int tid = threadIdx.x;
  for (int idx = tid; idx < 32 * NF; idx += 256) {
    const int r = idx >> 8, c = idx & 255;
    xs[r][c] = x[((size_t)(mt * 32 + r)) * NF + c];
  }
  __syncthreads();

  const int wave = tid >> 5;      // N tile 0..7
  const int lane = tid & 31;
  const int m  = lane & 15;
  const int hi = lane >> 4;       // 0: K pair {0,1}; 1: K pair {2,3}
  const int n  = wave * 16 + m;

  v8f c0 = {}, c1 = {};

  // Phase 1: x^2 columns (f = 0..255), Bmat rows 0..255
#pragma unroll 8
  for (int kk = 0; kk < 64; ++kk) {
    const int kb = kk * 4 + hi * 2;
    v2f bb; bb[0] = Bmat[kb * KST + n]; bb[1] = Bmat[(kb + 1) * KST + n];
    const float t0 = xs[m][kb],      t1 = xs[m][kb + 1];
    const float u0 = xs[16 + m][kb], u1 = xs[16 + m][kb + 1];
    v2f a0; a0[0] = t0 * t0; a0[1] = t1 * t1;
    v2f a1; a1[0] = u0 * u0; a1[1] = u1 * u1;
    c0 = __builtin_amdgcn_wmma_f32_16x16x4_f32(false, a0, false, bb, (short)0, c0, false, false);
    c1 = __builtin_amdgcn_wmma_f32_16x16x4_f32(false, a1, false, bb, (short)0, c1, false, false);
  }
  // Phase 2: x columns (f = 0..255), Bmat rows 256..511
#pragma unroll 8
  for (int kk = 0; kk < 64; ++kk) {
    const int kb = kk * 4 + hi * 2;
    v2f bb; bb[0] = Bmat[(NF + kb) * KST + n]; bb[1] = Bmat[(NF + kb + 1) * KST + n];
    v2f a0; a0[0] = xs[m][kb];      a0[1] = xs[m][kb + 1];
    v2f a1; a1[0] = xs[16 + m][kb]; a1[1] = xs[16 + m][kb + 1];
    c0 = __builtin_amdgcn_wmma_f32_16x16x4_f32(false, a0, false, bb, (short)0, c0, false, false);
    c1 = __builtin_amdgcn_wmma_f32_16x16x4_f32(false, a1, false, bb, (short)0, c1, false, false);
  }

  const float dn = dvec[n];
#pragma unroll
  for (int r = 0; r < 8; ++r) {
    const int gm0 = mt * 32 + r + hi * 8;
    logb[(size_t)gm0 * KST + n] = c0[r] + dn;
    logb[(size_t)(gm0 + 16) * KST + n] = c1[r] + dn;
  }
}

// ---------------------------------------------------------------------------
// Kernel 6: in-place cumsum over T per (b,k). Loads are independent of the
// running sum; unroll lets the compiler batch them ahead of the serial adds.
// ---------------------------------------------------------------------------
__global__ __launch_bounds__(KST) void cumsum_kernel(float* __restrict__ logb) {
  const int b = blockIdx.x, k = threadIdx.x;
  const size_t base = (size_t)b * TLEN * KST + k;
  float acc = 0.f;
#pragma unroll 8
  for (int t = 0; t < TLEN; ++t) {
    acc += logb[base + (size_t)t * KST];
    logb[base + (size_t)t * KST] = acc;
  }
}

// ---------------------------------------------------------------------------
// Kernel 7: sequential HSMM forward scan. One block per batch (16 WGs),
// 256 threads = (k in 0..127) x (half owning 24 durations). Ring buffers in
// LDS; cross-state reductions use wave32 shuffles (waves 0-3 hold exactly
// k=0..127) to minimize per-step barriers on the latency-critical chain.
// ---------------------------------------------------------------------------
__global__ __launch_bounds__(256) void hsmm_scan_kernel(
    const float* __restrict__ Cc, const float* __restrict__ Amat,
    const float* __restrict__ logDT, const float* __restrict__ logpi,
    float* __restrict__ out) {
  __shared__ float Mring[DMAX][KST];   // 24 KB
  __shared__ float Cring[DMAX][KST];   // 24 KB
  __shared__ float alpha_s[KST], p_s[KST];
  __shared__ float hm[2][KST], hs[2][KST], gp[2][KST];
  __shared__ float red[8];

  const int b = blockIdx.x, tid = threadIdx.x;
  const int k = tid & (KST - 1), half = tid >> 7;

  // init: slot 47 <-> time -1 carries logpi / C=0; older slots NEG / 0
  for (int idx = tid; idx < DMAX * KST; idx += 256) {
    const int d = idx / KST, kk = idx % KST;
    Mring[d][kk] = (d == DMAX - 1) ? logpi[kk] : NEGV;
    Cring[d][kk] = 0.f;
  }
  __syncthreads();

  const float* Cb = Cc + (size_t)b * TLEN * KST;
  float maxA = 0.f, p_last = 0.f;

  for (int t = 0; t < TLEN; ++t) {
    const float Ct = Cb[(size_t)t * KST + k];
    const int base = t % DMAX;

    // duration logsumexp: this thread owns 24 of the 48 durations
    float term[24];
    float hmax = NEGV;
#pragma unroll
    for (int i = 0; i < 24; ++i) {
      const int d = half * 24 + i + 1;
      int s = base - d; if (s < 0) s += DMAX;   // slot of time t-d
      term[i] = Ct - Cring[s][k] + logDT[(d - 1) * KST + k] + Mring[s][k];
      hmax = fmaxf(hmax, term[i]);
    }
    float hsum = 0.f;
#pragma unroll
    for (int i = 0; i < 24; ++i) hsum += expf(term[i] - hmax);
    hm[half][k] = hmax; hs[half][k] = hsum;
    __syncthreads();                                        // (1)
    if (half == 0) {
      const float m0 = hm[0][k], m1 = hm[1][k];
      const float mm = fmaxf(m0, m1);
      alpha_s[k] = mm + logf(hs[0][k] * expf(m0 - mm) + hs[1][k] * expf(m1 - mm));
    }
    __syncthreads();                                        // (2)

    // max over states: wave shuffle reduce, 1 barrier
    const float av = alpha_s[k];
    float v = av;
#pragma unroll
    for (int off = 16; off > 0; off >>= 1) v = fmaxf(v, __shfl_xor(v, off, 32));
    if ((tid & 31) == 0) red[tid >> 5] = v;
    __syncthreads();                                        // (3)
    maxA = fmaxf(fmaxf(red[0], red[1]), fmaxf(red[2], red[3]));
    p_last = expf(av - maxA);
    if (half == 0) p_s[k] = p_last;
    __syncthreads();                                        // (4)

    // message GEMV in probability domain (A = softmax rows, L2-resident)
    float g = 0.f;
    const float* Ar = Amat + (size_t)(half * 64) * KST + k;
#pragma unroll 8
    for (int j = 0; j < 64; ++j) g += p_s[half * 64 + j] * Ar[(size_t)j * KST];
    gp[half][k] = g;
    __syncthreads();                                        // (5)
    if (half == 0) {
      Mring[base][k] = maxA + logf(gp[0][k] + gp[1][k]);
      Cring[base][k] = Ct;
    }
    __syncthreads();                                        // (6)
  }

  // loglik = maxA + log(sum_k p): waves 0-3 hold k=0..127 exactly once
  float v = p_last;
#pragma unroll
  for (int off = 16; off > 0; off >>= 1) v += __shfl_xor(v, off, 32);
  if ((tid & 31) == 0) red[tid >> 5] = v;
  __syncthreads();
  if (tid == 0) out[b] = maxA + logf(red[0] + red[1] + red[2] + red[3]);
}

// ---------------------------------------------------------------------------
// Host launcher
// ---------------------------------------------------------------------------
extern "C" void kernel_launch(void* const* d_in, const int* in_sizes, int n_in,
                              void* d_out, int out_size, void* d_ws, size_t ws_size,
                              hipStream_t stream) {
  (void)in_sizes; (void)n_in; (void)out_size; (void)ws_size;
  const float* x            = (const float*)d_in[0];
  const float* context      = (const float*)d_in[1];
  const float* trans_logits = (const float*)d_in[2];
  const float* dur_logits   = (const float*)d_in[3];
  const float* mu           = (const float*)d_in[4];
  const float* log_var      = (const float*)d_in[5];
  const float* pi_logits    = (const float*)d_in[6];
  const float* ctx_A_w      = (const float*)d_in[7];
  const float* ctx_A_b      = (const float*)d_in[8];
  const float* ctx_D_w      = (const float*)d_in[9];
  const float* ctx_D_b      = (const float*)d_in[10];
  const float* ctx_E_w      = (const float*)d_in[11];
  const float* ctx_E_b      = (const float*)d_in[12];
  float* out = (float*)d_out;

  float* ws    = (float*)d_ws;
  float* Amat  = ws;                       // KST*KST          = 16384
  float* logDT = Amat  + KST * KST;        // DMAX*KST         = 6144
  float* Bmat  = logDT + DMAX * KST;       // 2*NF*KST         = 65536
  float* dvec  = Bmat  + 2 * NF * KST;     // KST              = 128
  float* lpi   = dvec  + KST;              // KST              = 128
  float* logb  = lpi   + KST;              // BATCH*TLEN*KST   = 1572864 (cumsum'd in place)

  ctx_transA_kernel<<<KST, KST, 0, stream>>>(trans_logits, ctx_A_w, ctx_A_b, context, Amat);
  ctx_durD_kernel  <<<KST, 64,  0, stream>>>(dur_logits, ctx_D_w, ctx_D_b, context, logDT);
  ctx_emis_kernel  <<<KST, NF,  0, stream>>>(mu, log_var, ctx_E_w, ctx_E_b, context, Bmat, dvec);
  pi_kernel        <<<1,   KST, 0, stream>>>(pi_logits, lpi);

  emis_gemm_kernel <<<(BATCH * TLEN) / 32, 256, 0, stream>>>(x, Bmat, dvec, logb);
  cumsum_kernel    <<<BATCH, KST, 0, stream>>>(logb);
  hsmm_scan_kernel <<<BATCH, 256, 0, stream>>>(logb, Amat, logDT, lpi, out);
}